// Net_imp_71768903516551
// MI455X (gfx1250) — compile-verified
//
#include <hip/hip_runtime.h>
#include <math.h>

typedef __attribute__((ext_vector_type(2))) float v2f;
typedef __attribute__((ext_vector_type(8))) float v8f;

#define HEADS 8
#define LRELU 0.2f

// ---------- ordered-uint encoding so float max can use native u32 atomicMax ----------
__device__ __forceinline__ unsigned ordf(float f) {
  unsigned u = __float_as_uint(f);
  return (u & 0x80000000u) ? ~u : (u | 0x80000000u);
}
__device__ __forceinline__ float deordf(unsigned u) {
  return __uint_as_float((u & 0x80000000u) ? (u ^ 0x80000000u) : ~u);
}

// ---------- WMMA fp32 GEMM: Y[M,Fout] = X[M,K] @ W[K,Fout] ----------
// grid.x = M/16 (M multiple of 16); blockDim.x = 32 * (Fout/16); one wave per 16x16 tile.
__global__ void wmma_gemm_f32(const float* __restrict__ X, const float* __restrict__ W,
                              float* __restrict__ Y, int K, int Fout) {
  const int wave = threadIdx.x >> 5;
  const int lane = threadIdx.x & 31;
  const int m0 = blockIdx.x << 4;
  const int n0 = wave << 4;
  const int r    = lane & 15;        // A row / B col (within tile)
  const int koff = (lane >> 4) << 1; // lanes 16-31 hold K+2,K+3
  v8f acc = {};
  const float* xrow = X + (size_t)(m0 + r) * K + koff;
  #pragma unroll 4
  for (int k = 0; k < K; k += 4) {
    v2f a, b;
    a.x = xrow[k];
    a.y = xrow[k + 1];
    const float* wp = W + (size_t)(k + koff) * Fout + n0 + r;
    b.x = wp[0];
    b.y = wp[Fout];
    acc = __builtin_amdgcn_wmma_f32_16x16x4_f32(false, a, false, b, (short)0, acc,
                                                false, false);
  }
  const int orow = (lane >> 4) << 3;   // +8 for upper half-lanes
  #pragma unroll
  for (int i = 0; i < 8; ++i)
    Y[(size_t)(m0 + orow + i) * Fout + n0 + r] = acc[i];
}

// ---------- per-node attention logits: al[n,h] = sum_c h[n,h,c] * a[h,c] ----------
__global__ void attn_logits(const float* __restrict__ h, const float* __restrict__ a_src,
                            const float* __restrict__ a_dst, float* __restrict__ al_s,
                            float* __restrict__ al_d, int N, int C) {
  int t = blockIdx.x * blockDim.x + threadIdx.x;
  if (t >= N * HEADS) return;
  int hh = t & (HEADS - 1);
  const float* hp = h + (size_t)t * C;  // t = n*HEADS + hh, layout [N, H*C] contiguous
  float ss = 0.f, sd = 0.f;
  for (int c = 0; c < C; ++c) {
    float v = hp[c];
    ss += v * a_src[hh * C + c];
    sd += v * a_dst[hh * C + c];
  }
  al_s[t] = ss;
  al_d[t] = sd;
}

// ---------- edge pass 1: segment max of leaky_relu(al_s[src]+al_d[dst]) ----------
__global__ void edge_max_k(const int* __restrict__ ei, int E, int N,
                           const float* __restrict__ al_s, const float* __restrict__ al_d,
                           unsigned* __restrict__ emax) {
  int t = blockIdx.x * blockDim.x + threadIdx.x;
  int total = (E + N) * HEADS;
  if (t >= total) return;
  int e = t >> 3, hh = t & 7;
  int s, d;
  if (e < E) { s = ei[e]; d = ei[E + e]; } else { s = d = e - E; }
  float v = al_s[s * HEADS + hh] + al_d[d * HEADS + hh];
  v = (v > 0.f) ? v : v * LRELU;
  atomicMax(&emax[d * HEADS + hh], ordf(v));
}

// ---------- edge pass 2: denom[dst,h] += exp(e - emax[dst,h]) ----------
__global__ void edge_sum_k(const int* __restrict__ ei, int E, int N,
                           const float* __restrict__ al_s, const float* __restrict__ al_d,
                           const unsigned* __restrict__ emax, float* __restrict__ denom) {
  int t = blockIdx.x * blockDim.x + threadIdx.x;
  int total = (E + N) * HEADS;
  if (t >= total) return;
  int e = t >> 3, hh = t & 7;
  int s, d;
  if (e < E) { s = ei[e]; d = ei[E + e]; } else { s = d = e - E; }
  float v = al_s[s * HEADS + hh] + al_d[d * HEADS + hh];
  v = (v > 0.f) ? v : v * LRELU;
  float ex = __expf(v - deordf(emax[d * HEADS + hh]));
  atomicAdd(&denom[d * HEADS + hh], ex);
}

// ---------- edge pass 3: agg[dst,h,:] += alpha * h[src,h,:] ----------
template <int C>
__global__ void edge_agg_k(const int* __restrict__ ei, int E, int N,
                           const float* __restrict__ al_s, const float* __restrict__ al_d,
                           const unsigned* __restrict__ emax, const float* __restrict__ denom,
                           const float* __restrict__ h, float* __restrict__ agg) {
  int t = blockIdx.x * blockDim.x + threadIdx.x;
  int total = (E + N) * HEADS;
  if (t >= total) return;
  int e = t >> 3, hh = t & 7;
  int s, d;
  if (e < E) { s = ei[e]; d = ei[E + e]; } else { s = d = e - E; }
  int di = d * HEADS + hh;
  float v = al_s[s * HEADS + hh] + al_d[di];
  v = (v > 0.f) ? v : v * LRELU;
  float alpha = __expf(v - deordf(emax[di])) / (denom[di] + 1e-16f);
  const float* hp = h + ((size_t)s * HEADS + hh) * C;
  float* op = agg + ((size_t)di) * C;
  #pragma unroll
  for (int c = 0; c < C; ++c) atomicAdd(&op[c], hp[c] * alpha);
}

// ---------- bias + ELU (in place) ----------
__global__ void bias_elu_k(float* __restrict__ buf, const float* __restrict__ bias,
                           int total, int F) {
  int t = blockIdx.x * blockDim.x + threadIdx.x;
  if (t >= total) return;
  float v = buf[t] + bias[t % F];
  buf[t] = (v > 0.f) ? v : (__expf(v) - 1.0f);
}

// ---------- collapse v_proj -> out_proj -> fc into one 16x32 matrix + bias ----------
__global__ void combine_proj_k(const float* __restrict__ in_w, const float* __restrict__ in_b,
                               const float* __restrict__ out_w, const float* __restrict__ out_b,
                               const float* __restrict__ fc_w, const float* __restrict__ fc_b,
                               float* __restrict__ Wc, float* __restrict__ cb) {
  int t = threadIdx.x;            // 512 threads: (l,i) = (t>>5, t&31)
  int l = t >> 5, i = t & 31;
  const float* Wv = in_w + 64 * 32;   // rows 64..95 of in_proj
  const float* bv = in_b + 64;
  float s = 0.f;
  for (int j = 0; j < 32; ++j) {
    float wo_wv = 0.f;
    for (int k = 0; k < 32; ++k) wo_wv += out_w[j * 32 + k] * Wv[k * 32 + i];
    s += fc_w[l * 32 + j] * wo_wv;
  }
  Wc[l * 32 + i] = s;
  if (i == 0) {
    float b = fc_b[l];
    for (int j = 0; j < 32; ++j) {
      float tb = out_b[j];
      for (int k = 0; k < 32; ++k) tb += out_w[j * 32 + k] * bv[k];
      b += fc_w[l * 32 + j] * tb;
    }
    cb[l] = b;
  }
}

// ---------- logits = h @ Wc^T + cb ; log_softmax over 16 ----------
__global__ void final_k(const float* __restrict__ h, const float* __restrict__ Wc,
                        const float* __restrict__ cb, float* __restrict__ out, int N) {
  __shared__ float sW[16 * 32];
  __shared__ float sb[16];
  for (int i = threadIdx.x; i < 512; i += blockDim.x) sW[i] = Wc[i];
  if (threadIdx.x < 16) sb[threadIdx.x] = cb[threadIdx.x];
  __syncthreads();
  int n = blockIdx.x * blockDim.x + threadIdx.x;
  if (n >= N) return;
  float hv[32];
  const float* hp = h + (size_t)n * 32;
  #pragma unroll
  for (int i = 0; i < 32; ++i) hv[i] = hp[i];
  float lg[16];
  float m = -INFINITY;
  #pragma unroll
  for (int l = 0; l < 16; ++l) {
    float s = sb[l];
    for (int i = 0; i < 32; ++i) s += hv[i] * sW[l * 32 + i];
    lg[l] = s;
    m = fmaxf(m, s);
  }
  float se = 0.f;
  #pragma unroll
  for (int l = 0; l < 16; ++l) se += __expf(lg[l] - m);
  float lse = __logf(se) + m;
  float* op = out + (size_t)n * 16;
  #pragma unroll
  for (int l = 0; l < 16; ++l) op[l] = lg[l] - lse;
}

extern "C" void kernel_launch(void* const* d_in, const int* in_sizes, int n_in,
                              void* d_out, int out_size, void* d_ws, size_t ws_size,
                              hipStream_t stream) {
  const float* x       = (const float*)d_in[0];
  const int*   ei      = (const int*)d_in[1];
  const float* W1      = (const float*)d_in[2];
  const float* a1_src  = (const float*)d_in[3];
  const float* a1_dst  = (const float*)d_in[4];
  const float* b1      = (const float*)d_in[5];
  const float* W2      = (const float*)d_in[6];
  const float* a2_src  = (const float*)d_in[7];
  const float* a2_dst  = (const float*)d_in[8];
  const float* b2      = (const float*)d_in[9];
  const float* in_w    = (const float*)d_in[10];
  const float* in_b    = (const float*)d_in[11];
  const float* out_w   = (const float*)d_in[12];
  const float* out_b   = (const float*)d_in[13];
  const float* fc_w    = (const float*)d_in[14];
  const float* fc_b    = (const float*)d_in[15];
  float* out = (float*)d_out;

  const int N = in_sizes[0] / 128;   // 50000 (multiple of 16)
  const int E = in_sizes[1] / 2;     // 1600000

  // workspace carve-out
  char* ws = (char*)d_ws;
  size_t off = 0;
  auto carve = [&](size_t nfloats) { float* p = (float*)(ws + off); off += nfloats * 4; return p; };
  float*    h1    = carve((size_t)N * 64);  // reused by layer 2 (h2 | agg2)
  float*    agg1  = carve((size_t)N * 64);
  float*    al_s  = carve((size_t)N * 8);
  float*    al_d  = carve((size_t)N * 8);
  unsigned* emax  = (unsigned*)carve((size_t)N * 8);
  float*    denom = carve((size_t)N * 8);
  float*    Wc    = carve(512);
  float*    cb    = carve(16);
  float*    h2    = h1;
  float*    agg2  = h1 + (size_t)N * 32;

  const int mtiles = N / 16;
  const int eth = (E + N) * HEADS;
  const int eblk = (eth + 255) / 256;

  // ===== GAT layer 1: 128 -> 8x8 =====
  wmma_gemm_f32<<<mtiles, 128, 0, stream>>>(x, W1, h1, 128, 64);
  attn_logits<<<(N * HEADS + 255) / 256, 256, 0, stream>>>(h1, a1_src, a1_dst, al_s, al_d, N, 8);
  hipMemsetAsync(emax, 0, (size_t)N * 8 * 4, stream);
  hipMemsetAsync(denom, 0, (size_t)N * 8 * 4, stream);
  hipMemsetAsync(agg1, 0, (size_t)N * 64 * 4, stream);
  edge_max_k<<<eblk, 256, 0, stream>>>(ei, E, N, al_s, al_d, emax);
  edge_sum_k<<<eblk, 256, 0, stream>>>(ei, E, N, al_s, al_d, emax, denom);
  edge_agg_k<8><<<eblk, 256, 0, stream>>>(ei, E, N, al_s, al_d, emax, denom, h1, agg1);
  bias_elu_k<<<((size_t)N * 64 + 255) / 256, 256, 0, stream>>>(agg1, b1, N * 64, 64);

  // ===== GAT layer 2: 64 -> 8x4 =====
  wmma_gemm_f32<<<mtiles, 64, 0, stream>>>(agg1, W2, h2, 64, 32);
  attn_logits<<<(N * HEADS + 255) / 256, 256, 0, stream>>>(h2, a2_src, a2_dst, al_s, al_d, N, 4);
  hipMemsetAsync(emax, 0, (size_t)N * 8 * 4, stream);
  hipMemsetAsync(denom, 0, (size_t)N * 8 * 4, stream);
  hipMemsetAsync(agg2, 0, (size_t)N * 32 * 4, stream);
  edge_max_k<<<eblk, 256, 0, stream>>>(ei, E, N, al_s, al_d, emax);
  edge_sum_k<<<eblk, 256, 0, stream>>>(ei, E, N, al_s, al_d, emax, denom);
  edge_agg_k<4><<<eblk, 256, 0, stream>>>(ei, E, N, al_s, al_d, emax, denom, h2, agg2);
  bias_elu_k<<<((size_t)N * 32 + 255) / 256, 256, 0, stream>>>(agg2, b2, N * 32, 32);

  // ===== collapsed MHA(v,out_proj) + fc + log_softmax =====
  combine_proj_k<<<1, 512, 0, stream>>>(in_w, in_b, out_w, out_b, fc_w, fc_b, Wc, cb);
  final_k<<<(N + 255) / 256, 256, 0, stream>>>(agg2, Wc, cb, out, N);
}